// GCN_Decoder2_9732395893188
// MI455X (gfx1250) — compile-verified
//
#include <hip/hip_runtime.h>
#include <hip/hip_bf16.h>
#include <math.h>
#include <stdint.h>

// ---------------------------------------------------------------------------
// GCNII (GCN2Conv x4) + FC/BN/FC head, fp32, for MI455X (gfx1250, wave32).
//
// N=40000 nodes, D=128, E=640000 edges. Feature matrix (20.5MB) + edge list
// (10MB) are L2-resident (192MB L2) -> SpMM is L2-atomic bound. Dense layers
// use native fp32 WMMA (v_wmma_f32_16x16x4_f32) with:
//   - 64x16 output tile per wave (4 acc, B fragment shared by 4 WMMAs)
//   - W panel staged in LDS (64KB, k-tiled by 128), B frags via ds_load
//
// Workspace layout (d_ws), ~82.1 MB required:
//   norm  : N floats                (160000 B)
//   h     : N*D floats              (20.48 MB)
//   feat  : N*D floats (agg/feat)   (20.48 MB)
//   h2    : N*2D floats             (40.96 MB)
//   stats : 2*2D floats             (2 KB)
// ---------------------------------------------------------------------------

#define N_NODES 40000
#define N_EDGES 640000
#define DIM     128
#define DIM2    256
#define NLAYERS 4
#define ALPHA_C 0.1f
#define EPS_C   1e-5f

typedef float v2f __attribute__((ext_vector_type(2)));
typedef float v8f __attribute__((ext_vector_type(8)));

// ------------------------------- utility -----------------------------------

__global__ void zero_kernel(float* __restrict__ p, int n) {
  int i = blockIdx.x * blockDim.x + threadIdx.x;
  if (i < n) p[i] = 0.0f;
}

// deg[dst] += 1 over all edges (into `norm` buffer)
__global__ void degree_kernel(const long long* __restrict__ dst,
                              float* __restrict__ deg, int e) {
  int i = blockIdx.x * blockDim.x + threadIdx.x;
  if (i < e) atomicAdd(&deg[(int)dst[i]], 1.0f);
}

// norm = rsqrt(max(deg, 1)) in place
__global__ void norm_kernel(float* __restrict__ deg, int n) {
  int i = blockIdx.x * blockDim.x + threadIdx.x;
  if (i < n) {
    float d = deg[i];
    d = d < 1.0f ? 1.0f : d;
    deg[i] = rsqrtf(d);
  }
}

// ------------------------------- SpMM --------------------------------------
// One wave per edge: 32 lanes x float4 = 128 features.
// agg[dst] += h[src] * norm[src]   (f32 atomics, L2-resident working set)
__global__ void spmm_kernel(const long long* __restrict__ src,
                            const long long* __restrict__ dst,
                            const float* __restrict__ h,
                            const float* __restrict__ norm,
                            float* __restrict__ agg) {
  int gtid = blockIdx.x * blockDim.x + threadIdx.x;
  int edge = gtid >> 5;
  int lane = threadIdx.x & 31;
  if (edge >= N_EDGES) return;
  int s = (int)src[edge];
  int d = (int)dst[edge];
  float ns = norm[s];
  const float4* hp = (const float4*)(h + (size_t)s * DIM);
  float4 v = hp[lane];
  float* ap = agg + (size_t)d * DIM + lane * 4;
  atomicAdd(ap + 0, v.x * ns);
  atomicAdd(ap + 1, v.y * ns);
  atomicAdd(ap + 2, v.z * ns);
  atomicAdd(ap + 3, v.w * ns);
}

// feat = (1-alpha) * agg * norm[row] + alpha * x0   (in place on agg buffer)
__global__ void feat_kernel(float* __restrict__ agg,
                            const float* __restrict__ x0,
                            const float* __restrict__ norm) {
  int i = blockIdx.x * blockDim.x + threadIdx.x;
  if (i < N_NODES * DIM) {
    int row = i >> 7;  // DIM == 128
    agg[i] = (1.0f - ALPHA_C) * agg[i] * norm[row] + ALPHA_C * x0[i];
  }
}

// ------------------------------- WMMA GEMM ---------------------------------
// D = A[M,K] @ W[K,NCOLS] with epilogue.
//   EP == 0 : out = (1-beta)*A[m][n] + beta*acc + bias[n]   (requires K==NCOLS)
//   EP == 1 : out = relu(acc + bias[n])
//
// Block = 256 threads = 8 waves. Block output tile = 64 rows x 128 cols.
// Wave output tile = 64 rows x 16 cols (4 accumulators share each B frag).
// W panel (k-tile 128 x 128 cols = 64KB) staged in LDS; B frags via ds_load.
// Grid = (M/64, NCOLS/128). M = 40000 = 625*64, K/NCOLS in {128,256}: exact.
//
// fp32 WMMA VGPR layouts (ISA 7.12.2):
//   A 16x4 : lane m = lane&15, k-pair base kb = 2*(lane>>4); a = {A[m][kb], A[m][kb+1]}
//   B 4x16 : lane n = lane&15, same kb;                      b = {B[kb][n], B[kb+1][n]}
//   C/D    : 8 VGPRs, element r -> row r + 8*(lane>>4), col lane&15
template <int K, int NCOLS, int EP>
__global__ void wmma_gemm_kernel(const float* __restrict__ A,
                                 const float* __restrict__ W,
                                 const float* __restrict__ bias,
                                 float* __restrict__ out,
                                 float beta_coef) {
  __shared__ float bsm[128 * 128];  // 64 KB k-tile x col-panel of W

  const int lane    = threadIdx.x & 31;
  const int wave    = threadIdx.x >> 5;
  const int rm      = lane & 15;          // A row within 16-row tile
  const int kb      = (lane >> 4) << 1;   // k-pair base: 0 or 2
  const int colBase = blockIdx.y * 128;   // global col of this block's panel
  const int ncol    = (wave << 4) + (lane & 15);  // local col 0..127
  const int m0      = blockIdx.x * 64;    // first row of block tile

  // Per-M-tile A row pointers (v2f: contiguous k-pair loads, k+kb is even)
  const v2f* ap0 = (const v2f*)(A + ((size_t)m0 +  0 + rm) * (size_t)K);
  const v2f* ap1 = (const v2f*)(A + ((size_t)m0 + 16 + rm) * (size_t)K);
  const v2f* ap2 = (const v2f*)(A + ((size_t)m0 + 32 + rm) * (size_t)K);
  const v2f* ap3 = (const v2f*)(A + ((size_t)m0 + 48 + rm) * (size_t)K);

  v8f acc0 = {}, acc1 = {}, acc2 = {}, acc3 = {};

  for (int k0 = 0; k0 < K; k0 += 128) {
    // Stage W[k0..k0+127][colBase..colBase+127] -> bsm (row-major, 128 cols).
    // 16384 floats / 256 threads = 16 float4 per thread, fully coalesced.
#pragma unroll
    for (int j = 0; j < 16; ++j) {
      int idx  = (threadIdx.x + j * 256) << 2;  // 0..16380, step 4
      int krow = idx >> 7;
      int col  = idx & 127;
      *(float4*)(bsm + idx) =
          *(const float4*)(W + (size_t)(k0 + krow) * NCOLS + colBase + col);
    }
    __syncthreads();

#pragma unroll
    for (int kk = 0; kk < 128; kk += 4) {
      v2f b;
      b.x = bsm[(kk + kb) * 128 + ncol];          // ds_load_b32
      b.y = bsm[(kk + kb + 1) * 128 + ncol];      // ds_load_b32
      const int kidx = (k0 + kk + kb) >> 1;
      v2f a0 = ap0[kidx];                          // global_load_b64 (contig)
      v2f a1 = ap1[kidx];
      v2f a2 = ap2[kidx];
      v2f a3 = ap3[kidx];
      acc0 = __builtin_amdgcn_wmma_f32_16x16x4_f32(false, a0, false, b,
                                                   (short)0, acc0, false, false);
      acc1 = __builtin_amdgcn_wmma_f32_16x16x4_f32(false, a1, false, b,
                                                   (short)0, acc1, false, false);
      acc2 = __builtin_amdgcn_wmma_f32_16x16x4_f32(false, a2, false, b,
                                                   (short)0, acc2, false, false);
      acc3 = __builtin_amdgcn_wmma_f32_16x16x4_f32(false, a3, false, b,
                                                   (short)0, acc3, false, false);
    }
    __syncthreads();
  }

  // Epilogue: 4 M-tiles of 16x16. D elem r -> row r + 8*(lane>>4), col lane&15.
  const int nglob = colBase + ncol;
  const float bv  = bias[nglob];
  const int rbase = ((lane >> 4) << 3);
  v8f accs[4] = {acc0, acc1, acc2, acc3};
#pragma unroll
  for (int mt = 0; mt < 4; ++mt) {
#pragma unroll
    for (int r = 0; r < 8; ++r) {
      const int m = m0 + (mt << 4) + rbase + r;
      float v = accs[mt][r];
      float o;
      if constexpr (EP == 0) {
        float fv = A[(size_t)m * (size_t)K + nglob];  // K == NCOLS here
        o = (1.0f - beta_coef) * fv + beta_coef * v + bv;
      } else {
        o = v + bv;
        o = o > 0.0f ? o : 0.0f;
      }
      out[(size_t)m * (size_t)NCOLS + nglob] = o;
    }
  }
}

// ------------------------------- BatchNorm ---------------------------------

#define BN_ROWS_PER_BLOCK 128

// stats[c] += sum, stats[DIM2+c] += sumsq over a row chunk. threadIdx.x = channel.
__global__ void bn_stats_kernel(const float* __restrict__ h2,
                                float* __restrict__ stats) {
  int c = threadIdx.x;  // 0..255
  int r0 = blockIdx.x * BN_ROWS_PER_BLOCK;
  int r1 = r0 + BN_ROWS_PER_BLOCK;
  if (r1 > N_NODES) r1 = N_NODES;
  float s = 0.0f, s2 = 0.0f;
  for (int r = r0; r < r1; ++r) {
    float v = h2[(size_t)r * DIM2 + c];
    s += v;
    s2 += v * v;
  }
  atomicAdd(&stats[c], s);
  atomicAdd(&stats[DIM2 + c], s2);
}

// stats -> {scale, shift}: scale = gamma*rsqrt(var+eps), shift = beta - mean*scale
__global__ void bn_final_kernel(float* __restrict__ stats,
                                const float* __restrict__ gamma,
                                const float* __restrict__ beta) {
  int c = threadIdx.x;  // one block of 256
  float mean = stats[c] * (1.0f / (float)N_NODES);
  float var  = stats[DIM2 + c] * (1.0f / (float)N_NODES) - mean * mean;
  float inv  = rsqrtf(var + EPS_C);
  float sc   = gamma[c] * inv;
  stats[c]        = sc;
  stats[DIM2 + c] = beta[c] - mean * sc;
}

__global__ void bn_apply_kernel(float* __restrict__ h2,
                                const float* __restrict__ stats) {
  int i = blockIdx.x * blockDim.x + threadIdx.x;
  if (i < N_NODES * DIM2) {
    int c = i & (DIM2 - 1);
    h2[i] = h2[i] * stats[c] + stats[DIM2 + c];
  }
}

// ------------------------------- launch ------------------------------------

extern "C" void kernel_launch(void* const* d_in, const int* in_sizes, int n_in,
                              void* d_out, int out_size, void* d_ws, size_t ws_size,
                              hipStream_t stream) {
  const float*     x        = (const float*)d_in[0];
  const float*     gnn_w    = (const float*)d_in[1];   // [L, D, D]
  const float*     gnn_b    = (const float*)d_in[2];   // [L, D]
  const float*     fc_w     = (const float*)d_in[3];   // [D, 2D]
  const float*     fc_b     = (const float*)d_in[4];   // [2D]
  const float*     bn_gamma = (const float*)d_in[5];   // [2D]
  const float*     bn_beta  = (const float*)d_in[6];   // [2D]
  const float*     fc2_w    = (const float*)d_in[7];   // [2D, D]
  const float*     fc2_b    = (const float*)d_in[8];   // [D]
  const long long* edge     = (const long long*)d_in[9];  // [2, E] int64
  const long long* esrc = edge;
  const long long* edst = edge + N_EDGES;

  char* ws = (char*)d_ws;
  float* norm  = (float*)ws;  ws += (size_t)N_NODES * 4;          // 160000 B (256-aligned)
  float* h     = (float*)ws;  ws += (size_t)N_NODES * DIM  * 4;
  float* feat  = (float*)ws;  ws += (size_t)N_NODES * DIM  * 4;
  float* h2    = (float*)ws;  ws += (size_t)N_NODES * DIM2 * 4;
  float* stats = (float*)ws;                                       // 2*DIM2 floats

  // --- degree / norm ---
  zero_kernel<<<(N_NODES + 255) / 256, 256, 0, stream>>>(norm, N_NODES);
  degree_kernel<<<(N_EDGES + 255) / 256, 256, 0, stream>>>(edst, norm, N_EDGES);
  norm_kernel<<<(N_NODES + 255) / 256, 256, 0, stream>>>(norm, N_NODES);

  // h = x
  hipMemcpyAsync(h, d_in[0], (size_t)N_NODES * DIM * sizeof(float),
                 hipMemcpyDeviceToDevice, stream);

  // --- GCNII layers ---
  const int spmm_blocks = (N_EDGES * 32) / 256;     // one wave per edge
  const int nd_blocks   = (N_NODES * DIM) / 256;
  for (int l = 0; l < NLAYERS; ++l) {
    float beta = logf(1.0f / (float)(l + 1) + 1.0f);  // LAMBDA = 1
    zero_kernel<<<nd_blocks, 256, 0, stream>>>(feat, N_NODES * DIM);
    spmm_kernel<<<spmm_blocks, 256, 0, stream>>>(esrc, edst, h, norm, feat);
    feat_kernel<<<nd_blocks, 256, 0, stream>>>(feat, x, norm);
    // h = (1-beta)*feat + beta*(feat @ W_l) + b_l
    wmma_gemm_kernel<DIM, DIM, 0><<<dim3(N_NODES / 64, 1), 256, 0, stream>>>(
        feat, gnn_w + (size_t)l * DIM * DIM, gnn_b + (size_t)l * DIM, h, beta);
  }

  // --- FC1 + ReLU ---
  wmma_gemm_kernel<DIM, DIM2, 1><<<dim3(N_NODES / 64, 2), 256, 0, stream>>>(
      h, fc_w, fc_b, h2, 0.0f);

  // --- BatchNorm (training-mode batch stats) ---
  zero_kernel<<<(2 * DIM2 + 255) / 256, 256, 0, stream>>>(stats, 2 * DIM2);
  bn_stats_kernel<<<(N_NODES + BN_ROWS_PER_BLOCK - 1) / BN_ROWS_PER_BLOCK, DIM2,
                    0, stream>>>(h2, stats);
  bn_final_kernel<<<1, DIM2, 0, stream>>>(stats, bn_gamma, bn_beta);
  bn_apply_kernel<<<(N_NODES * DIM2) / 256, 256, 0, stream>>>(h2, stats);

  // --- FC2 + ReLU -> d_out ---
  wmma_gemm_kernel<DIM2, DIM, 1><<<dim3(N_NODES / 64, 1), 256, 0, stream>>>(
      h2, fc2_w, fc2_b, (float*)d_out, 0.0f);
}